// SegSampling_49967649522154
// MI455X (gfx1250) — compile-verified
//
#include <hip/hip_runtime.h>
#include <math.h>

typedef float v2f __attribute__((ext_vector_type(2)));
typedef float v8f __attribute__((ext_vector_type(8)));

#define Q_TOT 4096
#define E_DIM 128
#define P_PTS 8
#define N_CAM 6
#define CF_CH 132
#define OFF_O 24   // G*P*3
#define SCL_O 32   // G*P*L

// workspace layout (float offsets)
#define WS_OFF   0          // off_act   [4096*24]
#define WS_SWRAW 98304      // sw_raw    [4096*32]
#define WS_PTS   229376     // pts       [32768*3]
#define WS_SWQ   327680     // softmaxed [32768*4]
#define WS_U     458752     // u         [32768*6]
#define WS_V     655360     // v         [32768*6]
#define WS_VAL   851968     // valid     [32768*6] (int)
#define WS_FEATT 1048576    // channel-last feats [2962080]

// output layout (float offsets)
#define O_FEAT   0
#define O_WFEAT  4194304
#define O_WGT    8388608
#define O_POS3D  8421376
#define O_UNC    8519680

// ---------------------------------------------------------------------------
// K1: fused head GEMM  raw[q,o] = sum_e query[e,q]*W[o,e] + b[o]
// One wave per 16x16 tile, V_WMMA_F32_16X16X4_F32, full fp32 precision.
// Padded lanes (o >= Osz) compute garbage columns that are never stored, so
// the B address is merely clamped in-bounds -- no masking, no divergence.
// nt<2 -> offset head (O=24, sigmoid applied); nt>=2 -> scale head (O=32, raw).
// ---------------------------------------------------------------------------
__global__ __launch_bounds__(32) void k1_gemm(
    const float* __restrict__ q,
    const float* __restrict__ offw, const float* __restrict__ offb,
    const float* __restrict__ sclw, const float* __restrict__ sclb,
    float* __restrict__ off_act, float* __restrict__ sw_raw)
{
  const int bm   = blockIdx.x;        // 256 M-tiles
  const int nt   = blockIdx.y;        // 4 N-tiles (2 offset + 2 scale)
  const int lane = threadIdx.x;
  const int half = lane >> 4;
  const int lm   = lane & 15;
  const int m0   = bm * 16;

  const float* Wm; const float* bias; int Osz, o0; float* dst;
  if (nt < 2) { Wm = offw; bias = offb; Osz = OFF_O; o0 = nt * 16;       dst = off_act; }
  else        { Wm = sclw; bias = sclb; Osz = SCL_O; o0 = (nt - 2) * 16; dst = sw_raw;  }

  const int  o  = o0 + lm;
  const bool ov = (o < Osz);
  const int  oc = ov ? o : (Osz - 1);            // clamp address only

  const float* __restrict__ Acol = q + m0 + lm;                  // column m0+lm
  const v2f*   __restrict__ Brow = (const v2f*)(Wm + (size_t)oc * E_DIM);

  v8f c = {};
#pragma unroll 8
  for (int k0 = 0; k0 < E_DIM; k0 += 4) {
    const int ka = k0 + 2 * half;      // lane<16 -> K=k0,k0+1 ; lane>=16 -> k0+2,k0+3
    v2f a;
    a.x = Acol[(size_t)ka       * Q_TOT];
    a.y = Acol[(size_t)(ka + 1) * Q_TOT];
    const v2f b = Brow[ka >> 1];       // contiguous pair -> global_load_b64
    c = __builtin_amdgcn_wmma_f32_16x16x4_f32(false, a, false, b, (short)0, c,
                                              false, false);
  }

  if (ov) {
    const float bv = bias[o];
#pragma unroll
    for (int r = 0; r < 8; ++r) {
      float val = c[r] + bv;                        // row M = r + 8*half
      if (nt < 2) val = 1.f / (1.f + expf(-val));   // sigmoid for offsets
      dst[(m0 + r + 8 * half) * Osz + o] = val;
    }
  }
}

// ---------------------------------------------------------------------------
// K2: per-(q,p) geometry: softmax over L, offset->world points, 6-cam project.
// ---------------------------------------------------------------------------
__global__ __launch_bounds__(256) void k2_geom(
    const float* __restrict__ refp, const float* __restrict__ l2i,
    const float* __restrict__ off_act, const float* __restrict__ sw_raw,
    float* __restrict__ pts, float* __restrict__ swq,
    float* __restrict__ ua, float* __restrict__ va, int* __restrict__ vld)
{
  const int t = blockIdx.x * blockDim.x + threadIdx.x;
  if (t >= Q_TOT * P_PTS) return;
  const int qi = t >> 3, p = t & 7;

  // softmax over L=4
  float s[4];
#pragma unroll
  for (int l = 0; l < 4; ++l) s[l] = sw_raw[qi * SCL_O + p * 4 + l];
  float m = fmaxf(fmaxf(s[0], s[1]), fmaxf(s[2], s[3]));
  float e[4], sum = 0.f;
#pragma unroll
  for (int l = 0; l < 4; ++l) { e[l] = expf(s[l] - m); sum += e[l]; }
  const float inv = 1.f / sum;
#pragma unroll
  for (int l = 0; l < 4; ++l) swq[t * 4 + l] = e[l] * inv;

  // offsets -> world points
  const float r  = 0.25f + 1e-6f;
  const float zr = 4.0f  + 1e-6f;
  const float ox = off_act[qi * OFF_O + p * 3 + 0] * (2.f * r)  - r;
  const float oy = off_act[qi * OFF_O + p * 3 + 1] * (2.f * r)  - r;
  const float oz = off_act[qi * OFF_O + p * 3 + 2] * (2.f * zr) - zr;
  const float px = refp[qi * 3 + 0] * 100.f - 50.f + ox;
  const float py = refp[qi * 3 + 1] * 100.f - 50.f + oy;
  const float pz = refp[qi * 3 + 2] * 8.f   - 5.f  + oz;
  pts[t * 3 + 0] = px; pts[t * 3 + 1] = py; pts[t * 3 + 2] = pz;

  for (int n = 0; n < N_CAM; ++n) {
    const float* M4 = l2i + n * 16;
    const float cx = M4[0] * px + M4[1] * py + M4[2]  * pz + M4[3];
    const float cy = M4[4] * px + M4[5] * py + M4[6]  * pz + M4[7];
    const float cz = M4[8] * px + M4[9] * py + M4[10] * pz + M4[11];
    const float zc = fmaxf(cz, 1e-5f);
    const float u  = cx / zc / 704.f;
    const float v  = cy / zc / 256.f;
    const int ok = (cz > 1e-5f) & (u > 0.f) & (u < 1.f) & (v > 0.f) & (v < 1.f);
    ua[t * N_CAM + n] = u;
    va[t * N_CAM + n] = v;
    vld[t * N_CAM + n] = ok;
  }
}

// ---------------------------------------------------------------------------
// K3: transpose feats [N,C,HW] -> channel-last [N,HW,C] (LDS-tiled 32x32).
// ---------------------------------------------------------------------------
__global__ __launch_bounds__(256) void k3_tr(
    const float* __restrict__ f0, const float* __restrict__ f1,
    const float* __restrict__ f2, const float* __restrict__ f3,
    float* __restrict__ featT)
{
  const int lw[4]    = {88, 44, 22, 11};
  const int lh[4]    = {32, 16, 8, 4};
  const int lbase[4] = {0, 2230272, 2787840, 2927232};
  const int z = blockIdx.z;
  const int n = z >> 2, l = z & 3;
  const int HW = lw[l] * lh[l];
  const int p0 = blockIdx.x * 32;
  if (p0 >= HW) return;
  const int c0 = blockIdx.y * 32;

  const float* src = (l == 0 ? f0 : l == 1 ? f1 : l == 2 ? f2 : f3)
                   + (size_t)n * CF_CH * HW;
  float* dst = featT + lbase[l] + (size_t)n * HW * CF_CH;

  __shared__ float tile[32][33];
  const int tx = threadIdx.x, ty = threadIdx.y;
#pragma unroll
  for (int r = 0; r < 4; ++r) {
    const int cc = c0 + ty + r * 8, pp = p0 + tx;
    if (cc < CF_CH && pp < HW) tile[ty + r * 8][tx] = src[(size_t)cc * HW + pp];
  }
  __syncthreads();
#pragma unroll
  for (int r = 0; r < 4; ++r) {
    const int pp = p0 + ty + r * 8, cc = c0 + tx;
    if (cc < CF_CH && pp < HW) dst[(size_t)pp * CF_CH + cc] = tile[tx][ty + r * 8];
  }
}

// ---------------------------------------------------------------------------
// K4: gather/blend. One block (128 thr = 4 wave32) per query; lane = channel.
// ---------------------------------------------------------------------------
__global__ __launch_bounds__(128) void k4_sample(
    const float* __restrict__ featT,
    const float* __restrict__ pts, const float* __restrict__ swq,
    const float* __restrict__ ua, const float* __restrict__ va,
    const int* __restrict__ vld, float* __restrict__ out)
{
  const int lw[4]    = {88, 44, 22, 11};
  const int lh[4]    = {32, 16, 8, 4};
  const int lbase[4] = {0, 2230272, 2787840, 2927232};
  const int qi  = blockIdx.x;
  const int tid = threadIdx.x;
  __shared__ float ex[4];

  for (int p = 0; p < P_PTS; ++p) {
    const int qp = qi * P_PTS + p;
    float acc0 = 0.f, acc1 = 0.f;   // acc0: channel tid; acc1: channel 128+tid (tid<4)
    float wlv[4];
#pragma unroll
    for (int l = 0; l < 4; ++l) wlv[l] = swq[qp * 4 + l];

    for (int n = 0; n < N_CAM; ++n) {
      if (!vld[qp * N_CAM + n]) continue;       // uniform per block
      const float u = ua[qp * N_CAM + n];
      const float v = va[qp * N_CAM + n];
#pragma unroll
      for (int l = 0; l < 4; ++l) {
        const int W = lw[l], H = lh[l], HW = W * H;
        const float x = u * (float)W - 0.5f;
        const float y = v * (float)H - 0.5f;
        const float fx0 = floorf(x), fy0 = floorf(y);
        const int   x0 = (int)fx0,  y0 = (int)fy0;
        const float wx = x - fx0,   wy = y - fy0;
        const float* base = featT + lbase[l] + (size_t)n * HW * CF_CH;
        const float cw[4] = { (1.f - wx) * (1.f - wy) * wlv[l],
                              wx * (1.f - wy) * wlv[l],
                              (1.f - wx) * wy * wlv[l],
                              wx * wy * wlv[l] };
        const int cxi[4] = { x0, x0 + 1, x0,     x0 + 1 };
        const int cyi[4] = { y0, y0,     y0 + 1, y0 + 1 };
#pragma unroll
        for (int k = 0; k < 4; ++k) {
          if ((unsigned)cxi[k] < (unsigned)W && (unsigned)cyi[k] < (unsigned)H) {
            const float* row = base + ((size_t)cyi[k] * W + cxi[k]) * CF_CH;
            acc0 += cw[k] * row[tid];
            if (tid < 4) acc1 += cw[k] * row[E_DIM + tid];
          }
        }
      }
    }

    if (tid < 4) ex[tid] = acc1;
    __syncthreads();
    const float dx = pts[qp * 3 + 0] - ex[0];
    const float dy = pts[qp * 3 + 1] - ex[1];
    const float dz = pts[qp * 3 + 2] - ex[2];
    const float wgt = expf(-0.1f * (dx * dx + dy * dy + dz * dz));

    out[O_FEAT  + (size_t)qp * E_DIM + tid] = acc0;
    out[O_WFEAT + (size_t)qp * E_DIM + tid] = acc0 * wgt;
    if (tid == 0) out[O_WGT + qp] = wgt;
    if (tid <  3) out[O_POS3D + (size_t)qp * 3 + tid] = ex[tid];
    if (tid == 0) out[O_UNC + qp] = ex[3];
    __syncthreads();
  }
}

// ---------------------------------------------------------------------------
extern "C" void kernel_launch(void* const* d_in, const int* in_sizes, int n_in,
                              void* d_out, int out_size, void* d_ws, size_t ws_size,
                              hipStream_t stream) {
  (void)in_sizes; (void)n_in; (void)out_size; (void)ws_size;
  const float* query = (const float*)d_in[0];
  const float* refp  = (const float*)d_in[1];
  const float* l2i   = (const float*)d_in[2];
  const float* offw  = (const float*)d_in[3];
  const float* offb  = (const float*)d_in[4];
  const float* sclw  = (const float*)d_in[5];
  const float* sclb  = (const float*)d_in[6];
  const float* f0    = (const float*)d_in[7];
  const float* f1    = (const float*)d_in[8];
  const float* f2    = (const float*)d_in[9];
  const float* f3    = (const float*)d_in[10];

  float* ws      = (float*)d_ws;
  float* out     = (float*)d_out;
  float* off_act = ws + WS_OFF;
  float* sw_raw  = ws + WS_SWRAW;
  float* pts     = ws + WS_PTS;
  float* swq     = ws + WS_SWQ;
  float* u       = ws + WS_U;
  float* v       = ws + WS_V;
  int*   vld     = (int*)(ws + WS_VAL);
  float* featT   = ws + WS_FEATT;

  k1_gemm<<<dim3(256, 4), dim3(32), 0, stream>>>(query, offw, offb, sclw, sclb,
                                                 off_act, sw_raw);
  k2_geom<<<dim3(128), dim3(256), 0, stream>>>(refp, l2i, off_act, sw_raw,
                                               pts, swq, u, v, vld);
  k3_tr<<<dim3(88, 5, 24), dim3(32, 8), 0, stream>>>(f0, f1, f2, f3, featT);
  k4_sample<<<dim3(Q_TOT), dim3(128), 0, stream>>>(featT, pts, swq, u, v, vld, out);
}